// EdgeDecoder_36756330119415
// MI455X (gfx1250) — compile-verified
//
#include <hip/hip_runtime.h>

#define HIDDEN 128
#define EDGES_PER_BLOCK 128
#define LDS_ROW_STRIDE 288   // 256 dwords + 32 pad: hi-half lanes hit banks 32..63
#define LDS_FLOATS (128 * LDS_ROW_STRIDE)

typedef float v2f __attribute__((ext_vector_type(2)));
typedef float v8f __attribute__((ext_vector_type(8)));

__global__ __launch_bounds__(256) void edge_decoder_wmma(
    const float* __restrict__ z_user, const float* __restrict__ z_item,
    const int* __restrict__ row_idx, const int* __restrict__ col_idx,
    const float* __restrict__ W1, const float* __restrict__ b1,
    const float* __restrict__ W2, const float* __restrict__ b2,
    float* __restrict__ out, int num_edges)
{
    extern __shared__ float ldsW1[];   // 128 * 288 floats = 144 KB

    const int tid = threadIdx.x;

    // Stage W1 [256,128] row-major into B-fragment layout:
    //   lds[(k>>1)*288 + n*2 + (k&1)] = W1[k][n]
    // Vectorized: read 4 cols of row-pair (2j, 2j+1), write 8 contiguous dwords.
    for (int idx = tid; idx < 128 * 32; idx += 256) {
        int j  = idx >> 5;          // row pair 0..127  (K rows 2j, 2j+1)
        int n4 = (idx & 31) * 4;    // col 0,4,...,124
        const float4 r0 = *(const float4*)(W1 + (2 * j)     * HIDDEN + n4);
        const float4 r1 = *(const float4*)(W1 + (2 * j + 1) * HIDDEN + n4);
        v2f* dst = (v2f*)(ldsW1 + j * LDS_ROW_STRIDE + n4 * 2);
        v2f p0 = {r0.x, r1.x};
        v2f p1 = {r0.y, r1.y};
        v2f p2 = {r0.z, r1.z};
        v2f p3 = {r0.w, r1.w};
        dst[0] = p0; dst[1] = p1; dst[2] = p2; dst[3] = p3;
    }
    __syncthreads();

    const int lane = tid & 31;
    const int wave = tid >> 5;
    const int m    = lane & 15;   // A row / B,C,D column index within 16-tile
    const int hi   = lane >> 4;   // lane half selects K pair

    const int edge_base = blockIdx.x * EDGES_PER_BLOCK + wave * 16;
    int edge   = edge_base + m;
    int eclamp = edge < num_edges ? edge : (num_edges - 1);

    const float* zu = z_user + (long)row_idx[eclamp] * HIDDEN;
    const float* zi = z_item + (long)col_idx[eclamp] * HIDDEN;

    v8f zero = {};
    v8f acc[8];
#pragma unroll
    for (int t = 0; t < 8; ++t) acc[t] = zero;

    const int koff = 2 * hi;   // A layout: lane holds A[m][2*hi], A[m][2*hi+1]

    // ---- Software-pipelined K loop over concat(zu, zi): 64 steps of K=4 ----
    v2f a_cur = *(const v2f*)(zu + koff);
    v2f b_cur[8];
    {
        const float* bp0 = ldsW1 + hi * LDS_ROW_STRIDE + m * 2;
#pragma unroll
        for (int t = 0; t < 8; ++t) b_cur[t] = *(const v2f*)(bp0 + t * 32);
    }

#pragma unroll 2
    for (int kk = 0; kk < 64; ++kk) {
        // Prefetch next k-step (clamped; last iter redundantly reloads 63).
        int kp = (kk + 1 < 64) ? kk + 1 : 63;
        const float* pa = (kp < 32) ? (zu + kp * 4) : (zi + (kp - 32) * 4);
        v2f a_nxt = *(const v2f*)(pa + koff);
        const float* bp = ldsW1 + (kp * 2 + hi) * LDS_ROW_STRIDE + m * 2;
        v2f b_nxt[8];
#pragma unroll
        for (int t = 0; t < 8; ++t) b_nxt[t] = *(const v2f*)(bp + t * 32);

        // 8 N-tiles share the same A fragment (loaded once per k-step).
#pragma unroll
        for (int t = 0; t < 8; ++t) {
            acc[t] = __builtin_amdgcn_wmma_f32_16x16x4_f32(
                false, a_cur, false, b_cur[t], (short)0, acc[t], false, false);
        }

        a_cur = a_nxt;
#pragma unroll
        for (int t = 0; t < 8; ++t) b_cur[t] = b_nxt[t];
    }

    // ---- Epilogue: h = relu(acc + b1); partial[r] = sum_n h*W2[n] ----
    const float b2v = b2[0];
    float partial[8] = {0.f, 0.f, 0.f, 0.f, 0.f, 0.f, 0.f, 0.f};
#pragma unroll
    for (int t = 0; t < 8; ++t) {
        float b1v = b1[t * 16 + m];
        float w2v = W2[t * 16 + m];
#pragma unroll
        for (int r = 0; r < 8; ++r) {
            float h = acc[t][r] + b1v;
            h = h > 0.f ? h : 0.f;
            partial[r] += h * w2v;
        }
    }

    // C/D layout: VGPR r holds row M=r (lanes 0-15) and M=r+8 (lanes 16-31).
    // Butterfly-reduce the 16 N-lanes within each half (masks < 16 stay in half).
#pragma unroll
    for (int r = 0; r < 8; ++r) {
        float s = partial[r];
        s += __shfl_xor(s, 1, 32);
        s += __shfl_xor(s, 2, 32);
        s += __shfl_xor(s, 4, 32);
        s += __shfl_xor(s, 8, 32);
        if (m == 0) {
            int e = edge_base + r + 8 * hi;
            if (e < num_edges) out[e] = s + b2v;
        }
    }
}

extern "C" void kernel_launch(void* const* d_in, const int* in_sizes, int n_in,
                              void* d_out, int out_size, void* d_ws, size_t ws_size,
                              hipStream_t stream) {
    const float* z_user  = (const float*)d_in[0];
    const float* z_item  = (const float*)d_in[1];
    const int*   row_idx = (const int*)  d_in[2];
    const int*   col_idx = (const int*)  d_in[3];
    const float* W1      = (const float*)d_in[4];
    const float* b1      = (const float*)d_in[5];
    const float* W2      = (const float*)d_in[6];
    const float* b2      = (const float*)d_in[7];
    float*       out     = (float*)d_out;

    int num_edges = in_sizes[2];
    int blocks = (num_edges + EDGES_PER_BLOCK - 1) / EDGES_PER_BLOCK;
    size_t lds_bytes = LDS_FLOATS * sizeof(float);   // 147456 B

    // Allow >64KB dynamic LDS (no stream interaction; graph-capture safe).
    (void)hipFuncSetAttribute((const void*)edge_decoder_wmma,
                              hipFuncAttributeMaxDynamicSharedMemorySize,
                              (int)lds_bytes);

    edge_decoder_wmma<<<blocks, 256, lds_bytes, stream>>>(
        z_user, z_item, row_idx, col_idx, W1, b1, W2, b2, out, num_edges);
}